// AttentionCircuit_9646496547057
// MI455X (gfx1250) — compile-verified
//
#include <hip/hip_runtime.h>
#include <hip/hip_bf16.h>
#include <math.h>

typedef float vf2 __attribute__((ext_vector_type(2)));
typedef float vf4 __attribute__((ext_vector_type(4)));
typedef float v8f __attribute__((ext_vector_type(8)));

// ---------------------------------------------------------------------------
// Problem constants (from reference): B=2, S=2048, D=1024, P=2, H=16, dh=64
// Nqk=Nv=64 feature neurons, Nrqk=Nrv=32 restore neurons, r=16, ds=64.
// Tokens T = B*S = 4096.
// ---------------------------------------------------------------------------
#define T_TOK   4096
#define DIM     1024
#define NFEAT   64
#define RRANK   16
#define NREST   32

__device__ __forceinline__ v8f wmma4(vf2 a, vf2 b, v8f c) {
  // D = A(16x4,f32) * B(4x16,f32) + C(16x16,f32)
  return __builtin_amdgcn_wmma_f32_16x16x4_f32(false, a, false, b, (short)0, c,
                                               false, false);
}

// ---------------------------------------------------------------------------
// K0: repack f_neurons (n, d, r) -> W1 row-major (d, n*16+r), 1024 x 2048
// ---------------------------------------------------------------------------
__global__ __launch_bounds__(256) void repack_f_kernel(
    const float* __restrict__ fn, float* __restrict__ W1) {
  int idx = blockIdx.x * 256 + threadIdx.x;          // 0 .. 1024*2048-1
  if (idx >= DIM * 2048) return;
  int d = idx >> 11;
  int c = idx & 2047;
  int n = c >> 4;
  int r = c & 15;
  W1[idx] = fn[((size_t)n << 14) + (d << 4) + r];    // n*D*16 + d*16 + r
}

// ---------------------------------------------------------------------------
// Generic f32 WMMA GEMM: C(MxN) = A(MxK) * B(KxN).
// TRANSB=false: B row-major (K x N, ldb). TRANSB=true: B stored (N x K, ldb),
// i.e. C = A * B^T (used for W_O).
// Block tile 128x128, BK=16, 256 threads = 8 waves, 8 wmma tiles per wave.
// B staged K-pair-interleaved: sBp[kp][n][2] so a B fragment is one b64 load.
// Requires M%128==0, N%128==0, K%16==0 (true for all uses here).
// ---------------------------------------------------------------------------
template <bool TRANSB>
__global__ __launch_bounds__(256) void gemm_f32_wmma(
    const float* __restrict__ A, const float* __restrict__ B,
    float* __restrict__ C, int M, int N, int K, int lda, int ldb, int ldc) {
  __shared__ float sA[128 * 18];    // 128 rows x (16 + 2 pad)
  __shared__ float sBp[8 * 258];    // 8 K-pairs x (128 cols x 2 + 2 pad)
  const int tid  = threadIdx.x;
  const int lane = tid & 31;
  const int wid  = tid >> 5;
  const int wm   = wid & 3;        // 4 waves along M (32 rows each)
  const int wn   = wid >> 2;       // 2 waves along N (64 cols each)
  const int bm0  = blockIdx.y * 128;
  const int bn0  = blockIdx.x * 128;
  const int ml   = lane & 15;
  const int kh   = (lane >> 4) << 1;  // 0 or 2 (always even)

  const v8f zero = {0.f, 0.f, 0.f, 0.f, 0.f, 0.f, 0.f, 0.f};
  v8f acc[2][4];
#pragma unroll
  for (int i = 0; i < 2; ++i)
#pragma unroll
    for (int j = 0; j < 4; ++j) acc[i][j] = zero;

  for (int k0 = 0; k0 < K; k0 += 16) {
    // ---- stage A tile (128x16) ----
#pragma unroll
    for (int i = 0; i < 2; ++i) {
      int lin = tid + i * 256;                 // 0..511
      int row = lin >> 2;
      int c4  = (lin & 3) << 2;
      vf4 v = *(const vf4*)(A + (size_t)(bm0 + row) * lda + k0 + c4);
      float* d = &sA[row * 18 + c4];
      *(vf2*)d       = vf2{v[0], v[1]};
      *(vf2*)(d + 2) = vf2{v[2], v[3]};
    }
    // ---- stage B tile (16x128), K-pair interleaved ----
    if (!TRANSB) {
#pragma unroll
      for (int i = 0; i < 2; ++i) {
        int lin = tid + i * 256;
        int row = lin >> 5;                    // k within tile, 0..15
        int c4  = (lin & 31) << 2;             // n within tile
        vf4 v = *(const vf4*)(B + (size_t)(k0 + row) * ldb + bn0 + c4);
        float* d = &sBp[(row >> 1) * 258 + c4 * 2 + (row & 1)];
        d[0] = v[0]; d[2] = v[1]; d[4] = v[2]; d[6] = v[3];
      }
    } else {
#pragma unroll
      for (int i = 0; i < 2; ++i) {
        int lin = tid + i * 256;
        int n  = lin >> 2;                     // n within tile, 0..127
        int k4 = (lin & 3) << 2;               // k within tile
        vf4 v = *(const vf4*)(B + (size_t)(bn0 + n) * ldb + k0 + k4);
#pragma unroll
        for (int j = 0; j < 4; ++j) {
          int k = k4 + j;
          sBp[(k >> 1) * 258 + n * 2 + (k & 1)] = v[j];
        }
      }
    }
    __syncthreads();

#pragma unroll
    for (int kk = 0; kk < 16; kk += 4) {
      const int krow = kk + kh;                // even
      const float* brow = &sBp[(krow >> 1) * 258];
      vf2 af[2], bf[4];
#pragma unroll
      for (int i = 0; i < 2; ++i)
        af[i] = *(const vf2*)&sA[(32 * wm + 16 * i + ml) * 18 + krow];
#pragma unroll
      for (int j = 0; j < 4; ++j)
        bf[j] = *(const vf2*)&brow[(64 * wn + 16 * j + ml) * 2];
#pragma unroll
      for (int i = 0; i < 2; ++i)
#pragma unroll
        for (int j = 0; j < 4; ++j) acc[i][j] = wmma4(af[i], bf[j], acc[i][j]);
    }
    __syncthreads();
  }

  // ---- epilogue: C layout VGPR v -> row v + 8*(lane/16), col lane%16 ----
  const int mh = lane >> 4;
#pragma unroll
  for (int i = 0; i < 2; ++i)
#pragma unroll
    for (int j = 0; j < 4; ++j)
#pragma unroll
      for (int v = 0; v < 8; ++v) {
        int row = bm0 + 32 * wm + 16 * i + v + 8 * mh;
        int col = bn0 + 64 * wn + 16 * j + ml;
        C[(size_t)row * ldc + col] = acc[i][j][v];
      }
}

// ---------------------------------------------------------------------------
// K2: zero the f-sum accumulators (3 routes x 2 p x 32 restore-neurons)
// ---------------------------------------------------------------------------
__global__ void zero_f_kernel(float* __restrict__ f) {
  if (threadIdx.x < 192) f[threadIdx.x] = 0.f;
}

// ---------------------------------------------------------------------------
// K3: per-token routing. One wave per token.
//   h[p][r]   = sum_n all_h[t,n,r] * wfeat[p,t,n]     (butterfly reduce)
//   ctx[c]    = sum_n wfeat[p,t,n] * emb[n,c]
//   logits[m] = [ctx|h] . Wr[:,m]; w = softmax * gate
//   A[t, n*16+r] = sum_p h[p][r] * w[p][n]            (feeds restore GEMM)
//   fsum[route,p,n] += w   (aux statistic)
// ---------------------------------------------------------------------------
__device__ __forceinline__ void load16(const float* __restrict__ src,
                                       float a[16]) {
#pragma unroll
  for (int r4 = 0; r4 < 4; ++r4) {
    vf4 v = *(const vf4*)(src + r4 * 4);
#pragma unroll
    for (int j = 0; j < 4; ++j) a[r4 * 4 + j] = v[j];
  }
}

__device__ __forceinline__ void route_one(
    int lane, int t, const float* __restrict__ wfeat,
    const float* __restrict__ gate, const float a0[16], const float a1[16],
    const float* __restrict__ sEmb, const float* __restrict__ sWr,
    float* __restrict__ sFroute, float* __restrict__ Aout) {
  float wlane[2];
  float hh[2][16];
#pragma unroll
  for (int p = 0; p < 2; ++p) {
    const float* wf = wfeat + (size_t)p * (T_TOK * NFEAT) + (size_t)t * NFEAT;
    float f0 = wf[lane];
    float f1 = wf[lane + 32];
    float h[16];
#pragma unroll
    for (int r = 0; r < 16; ++r) h[r] = a0[r] * f0 + a1[r] * f1;
#pragma unroll
    for (int m = 16; m >= 1; m >>= 1)
#pragma unroll
      for (int r = 0; r < 16; ++r) h[r] += __shfl_xor(h[r], m, 32);

    // ctx: this lane owns columns c = lane and c = lane+32
    float c0 = 0.f, c1 = 0.f;
    for (int n = 0; n < 32; ++n) {
      float fn = __shfl(f0, n, 32);
      c0 += fn * sEmb[n * 64 + lane];
      c1 += fn * sEmb[n * 64 + lane + 32];
    }
    for (int n = 0; n < 32; ++n) {
      float fn = __shfl(f1, n, 32);
      c0 += fn * sEmb[(n + 32) * 64 + lane];
      c1 += fn * sEmb[(n + 32) * 64 + lane + 32];
    }
    // logits: lane m = lane (32 restore neurons)
    float lg = 0.f;
    for (int c = 0; c < 32; ++c) lg += __shfl(c0, c, 32) * sWr[c * 32 + lane];
    for (int c = 0; c < 32; ++c)
      lg += __shfl(c1, c, 32) * sWr[(c + 32) * 32 + lane];
#pragma unroll
    for (int r = 0; r < 16; ++r) lg += h[r] * sWr[(64 + r) * 32 + lane];
    // softmax over 32 lanes, gated
    float mx = lg;
#pragma unroll
    for (int m = 16; m >= 1; m >>= 1) mx = fmaxf(mx, __shfl_xor(mx, m, 32));
    float e = __expf(lg - mx);
    float s = e;
#pragma unroll
    for (int m = 16; m >= 1; m >>= 1) s += __shfl_xor(s, m, 32);
    float w = (e / s) * gate[(size_t)p * T_TOK + t];
    atomicAdd(&sFroute[p * 32 + lane], w);
    wlane[p] = w;
#pragma unroll
    for (int r = 0; r < 16; ++r) hh[p][r] = h[r];
  }
  float* dst = Aout + (size_t)t * (NREST * RRANK) + lane * RRANK;
#pragma unroll
  for (int r4 = 0; r4 < 4; ++r4) {
    vf4 v;
#pragma unroll
    for (int j = 0; j < 4; ++j)
      v[j] = hh[0][r4 * 4 + j] * wlane[0] + hh[1][r4 * 4 + j] * wlane[1];
    *(vf4*)(dst + r4 * 4) = v;
  }
}

__global__ __launch_bounds__(256) void route_kernel(
    const float* __restrict__ allh, const float* __restrict__ fqkQ,
    const float* __restrict__ fqkK, const float* __restrict__ fvw,
    const float* __restrict__ gQ, const float* __restrict__ gK,
    const float* __restrict__ gV, const float* __restrict__ embQK,
    const float* __restrict__ embV, const float* __restrict__ WrQK,
    const float* __restrict__ WrV, float* __restrict__ Aq,
    float* __restrict__ Ak, float* __restrict__ Av,
    float* __restrict__ fsum) {
  __shared__ float sEmbQK[4096], sEmbV[4096];
  __shared__ float sWrQK[2560], sWrV[2560];
  __shared__ float sF[192];
  const int tid = threadIdx.x, lane = tid & 31, wid = tid >> 5;
  for (int i = tid; i < 4096; i += 256) {
    sEmbQK[i] = embQK[i];
    sEmbV[i] = embV[i];
  }
  for (int i = tid; i < 2560; i += 256) {
    sWrQK[i] = WrQK[i];
    sWrV[i] = WrV[i];
  }
  if (tid < 192) sF[tid] = 0.f;
  __syncthreads();

  const int t = blockIdx.x * 8 + wid;
  const float* ah = allh + (size_t)t * 2048;   // [n(0..127)][r], qk then v
  float a0[16], a1[16];
  // QK feature activations: n = lane, lane+32
  load16(ah + (size_t)lane * 16, a0);
  load16(ah + (size_t)(lane + 32) * 16, a1);
  route_one(lane, t, fqkQ, gQ, a0, a1, sEmbQK, sWrQK, &sF[0], Aq);
  route_one(lane, t, fqkK, gK, a0, a1, sEmbQK, sWrQK, &sF[64], Ak);
  // V feature activations: n = 64+lane, 96+lane
  load16(ah + (size_t)(64 + lane) * 16, a0);
  load16(ah + (size_t)(96 + lane) * 16, a1);
  route_one(lane, t, fvw, gV, a0, a1, sEmbV, sWrV, &sF[128], Av);
  __syncthreads();
  if (tid < 192) atomicAdd(&fsum[tid], sF[tid]);
}

// ---------------------------------------------------------------------------
// K5: per-token q-norm scale: scale = (||Q|| > 1e-6) ? 1 : ||Q||*1e-6
// ---------------------------------------------------------------------------
__global__ __launch_bounds__(256) void qscale_kernel(
    const float* __restrict__ Q, float* __restrict__ scl) {
  const int lane = threadIdx.x & 31, wid = threadIdx.x >> 5;
  const int t = blockIdx.x * 8 + wid;
  const vf4* q = (const vf4*)(Q + (size_t)t * DIM);
  float acc = 0.f;
#pragma unroll
  for (int i = 0; i < 8; ++i) {
    vf4 v = q[i * 32 + lane];
    acc += v[0] * v[0] + v[1] * v[1] + v[2] * v[2] + v[3] * v[3];
  }
#pragma unroll
  for (int m = 16; m >= 1; m >>= 1) acc += __shfl_xor(acc, m, 32);
  if (lane == 0) {
    float n = sqrtf(acc);
    scl[t] = (n > 1e-6f) ? 1.f : n * 1e-6f;
  }
}

// ---------------------------------------------------------------------------
// K6: causal flash attention, f32 WMMA. Block = 64 queries x one (b,h).
// 256 threads (8 waves): wave (srow = wid%4, wn = wid>>2) owns output rows
// 16*srow..16*srow+15, dh cols 32*wn..32*wn+31 (two 16x16 C tiles in regs).
// K^T / V staged K-pair-interleaved so B fragments are single b64 loads.
// ---------------------------------------------------------------------------
__global__ __launch_bounds__(256) void attn_kernel(
    const float* __restrict__ Q, const float* __restrict__ Kt,
    const float* __restrict__ V, const float* __restrict__ scl,
    float* __restrict__ O) {
  __shared__ float sQ[64 * 66];
  __shared__ float sKVp[32 * 130];  // 32 K-pairs x (64 cols x 2 + 2 pad)
  __shared__ float sS[64 * 66];     // scores, then P
  __shared__ float sMax[64], sSum[64], sAlpha[64];
  const int tid = threadIdx.x, lane = tid & 31, wid = tid >> 5;
  const int srow = wid & 3, wn = wid >> 2;
  const int ml = lane & 15;
  const int kh = (lane >> 4) << 1;     // 0 or 2 (even)
  const int qb = blockIdx.x;           // query block 0..31
  const int bh = blockIdx.y;           // 0..31
  const int b = bh >> 4, h = bh & 15;
  const size_t tokQ0 = (size_t)b * 2048 + (size_t)qb * 64;
  const int col0 = h * 64;

  // stage Q, pre-scaled by 1/sqrt(dh) = 1/8
#pragma unroll
  for (int i = 0; i < 4; ++i) {
    int lin = tid + i * 256;           // 0..1023
    int row = lin >> 4;
    int c4 = (lin & 15) << 2;
    vf4 v = *(const vf4*)(Q + (tokQ0 + row) * DIM + col0 + c4);
    float* d = &sQ[row * 66 + c4];
    *(vf2*)d       = vf2{v[0] * 0.125f, v[1] * 0.125f};
    *(vf2*)(d + 2) = vf2{v[2] * 0.125f, v[3] * 0.125f};
  }
  if (tid < 64) {
    sMax[tid] = -1e30f;
    sSum[tid] = 0.f;
  }
  const v8f zero = {0.f, 0.f, 0.f, 0.f, 0.f, 0.f, 0.f, 0.f};
  v8f o0 = zero, o1 = zero;
  __syncthreads();

  for (int kb = 0; kb <= qb; ++kb) {
    const size_t tokK0 = (size_t)b * 2048 + (size_t)kb * 64;
    // ---- stage K^T: element (dh, key) -> sKVp[dh>>1][key*2 + (dh&1)] ----
#pragma unroll
    for (int i = 0; i < 4; ++i) {
      int lin = tid + i * 256;
      int key = lin >> 4;
      int d4 = (lin & 15) << 2;
      vf4 v = *(const vf4*)(Kt + (tokK0 + key) * DIM + col0 + d4);
#pragma unroll
      for (int j = 0; j < 4; ++j) {
        int dh = d4 + j;
        sKVp[(dh >> 1) * 130 + key * 2 + (dh & 1)] = v[j];
      }
    }
    __syncthreads();
    // ---- scores: S = Qs * K^T (contract dh=64) ----
    v8f s0 = zero, s1 = zero;
#pragma unroll
    for (int k0 = 0; k0 < 64; k0 += 4) {
      const int krow = k0 + kh;
      const float* brow = &sKVp[(krow >> 1) * 130];
      vf2 a  = *(const vf2*)&sQ[(16 * srow + ml) * 66 + krow];
      vf2 b0 = *(const vf2*)&brow[(32 * wn + ml) * 2];
      vf2 b1 = *(const vf2*)&brow[(32 * wn + 16 + ml) * 2];
      s0 = wmma4(a, b0, s0);
      s1 = wmma4(a, b1, s1);
    }
#pragma unroll
    for (int v = 0; v < 8; ++v) {
      int row = 16 * srow + v + 8 * (lane >> 4);
      sS[row * 66 + 32 * wn + ml] = s0[v];
      sS[row * 66 + 32 * wn + 16 + ml] = s1[v];
    }
    __syncthreads();
    // ---- online softmax (threads 0..63); V staged by everyone ----
    if (tid < 64) {
      int row = tid;
      int kmaxj = (qb * 64 + row) - kb * 64;   // unmasked j <= kmaxj
      float mold = sMax[row], mnew = mold;
      for (int j = 0; j < 64; ++j)
        if (j <= kmaxj) mnew = fmaxf(mnew, sS[row * 66 + j]);
      float alpha = __expf(mold - mnew);
      float rs = 0.f;
      for (int j = 0; j < 64; ++j) {
        float p = (j <= kmaxj) ? __expf(sS[row * 66 + j] - mnew) : 0.f;
        sS[row * 66 + j] = p;
        rs += p;
      }
      sMax[row] = mnew;
      sAlpha[row] = alpha;
      sSum[row] = sSum[row] * alpha + rs;
    }
    // ---- stage V: element (key, dh) -> sKVp[key>>1][dh*2 + (key&1)] ----
#pragma unroll
    for (int i = 0; i < 4; ++i) {
      int lin = tid + i * 256;
      int key = lin >> 4;
      int d4 = (lin & 15) << 2;
      vf4 v = *(const vf4*)(V + (tokK0 + key) * DIM + col0 + d4);
      float* d = &sKVp[(key >> 1) * 130 + d4 * 2 + (key & 1)];
      d[0] = v[0]; d[2] = v[1]; d[4] = v[2]; d[6] = v[3];
    }
    __syncthreads();
    // ---- rescale accumulators, then O += P * V (contract keys=64) ----
    float al[8];
#pragma unroll
    for (int v = 0; v < 8; ++v) al[v] = sAlpha[16 * srow + v + 8 * (lane >> 4)];
#pragma unroll
    for (int v = 0; v < 8; ++v) {
      o0[v] *= al[v];
      o1[v] *= al[v];
    }
#pragma unroll
    for (int k0 = 0; k0 < 64; k0 += 4) {
      const int krow = k0 + kh;
      const float* brow = &sKVp[(krow >> 1) * 130];
      vf2 a  = *(const vf2*)&sS[(16 * srow + ml) * 66 + krow];
      vf2 b0 = *(const vf2*)&brow[(32 * wn + ml) * 2];
      vf2 b1 = *(const vf2*)&brow[(32 * wn + 16 + ml) * 2];
      o0 = wmma4(a, b0, o0);
      o1 = wmma4(a, b1, o1);
    }
    __syncthreads();
  }
  // ---- epilogue: /l, apply q-norm scale, write out ----
#pragma unroll
  for (int v = 0; v < 8; ++v) {
    int row = 16 * srow + v + 8 * (lane >> 4);
    size_t tok = tokQ0 + row;
    float f = (1.f / sSum[row]) * scl[tok];
    O[tok * DIM + col0 + 32 * wn + ml] = o0[v] * f;
    O[tok * DIM + col0 + 32 * wn + 16 + ml] = o1[v] * f;
  }
}

// ---------------------------------------------------------------------------
// K8: aux = 32 * sum over (3 routes, p, n) of (fsum/4096)^2
// ---------------------------------------------------------------------------
__global__ void aux_kernel(const float* __restrict__ f,
                           float* __restrict__ out_aux) {
  if (threadIdx.x == 0) {
    float s = 0.f;
    for (int i = 0; i < 192; ++i) {
      float v = f[i] * (1.f / 4096.f);
      s += v * v;
    }
    out_aux[0] = 32.f * s;
  }
}

// ---------------------------------------------------------------------------
// Host side
// ---------------------------------------------------------------------------
extern "C" void kernel_launch(void* const* d_in, const int* in_sizes, int n_in,
                              void* d_out, int out_size, void* d_ws,
                              size_t ws_size, hipStream_t stream) {
  (void)in_sizes; (void)n_in; (void)ws_size;
  const float* x        = (const float*)d_in[0];
  const float* fqkQ     = (const float*)d_in[1];
  const float* fqkK     = (const float*)d_in[2];
  const float* fvw      = (const float*)d_in[3];
  const float* gQ       = (const float*)d_in[4];
  const float* gK       = (const float*)d_in[5];
  const float* gV       = (const float*)d_in[6];
  const float* fneur    = (const float*)d_in[7];
  const float* rneur    = (const float*)d_in[8];
  const float* fqk_emb  = (const float*)d_in[9];
  const float* fv_emb   = (const float*)d_in[10];
  const float* Wr_qk    = (const float*)d_in[11];
  const float* Wr_v     = (const float*)d_in[12];
  const float* W_O      = (const float*)d_in[13];
  float* out = (float*)d_out;
  float* ws  = (float*)d_ws;

  // workspace layout (float offsets)
  const size_t OFF_W1    = 0;                       // 1024*2048 = 2M floats
  const size_t OFF_ALLH  = 2097152;                 // 4096*2048 = 8M floats
  const size_t OFF_AQ    = OFF_ALLH + 8388608;      // 4096*512 each
  const size_t OFF_AK    = OFF_AQ + 2097152;
  const size_t OFF_AV    = OFF_AK + 2097152;
  const size_t OFF_Q     = OFF_AV + 2097152;        // 4096*1024 each
  const size_t OFF_K     = OFF_Q + 4194304;
  const size_t OFF_V     = OFF_K + 4194304;
  const size_t OFF_SCALE = OFF_V + 4194304;         // 4096
  const size_t OFF_F     = OFF_SCALE + 4096;        // 192
  const size_t OFF_ATTN  = OFF_ALLH;                // reuse (all_h dead)

  // 1) repack feature weights
  repack_f_kernel<<<8192, 256, 0, stream>>>(fneur, ws + OFF_W1);
  // 2) all_h = X(4096x1024) * W1(1024x2048)
  gemm_f32_wmma<false><<<dim3(16, 32), 256, 0, stream>>>(
      x, ws + OFF_W1, ws + OFF_ALLH, 4096, 2048, 1024, 1024, 2048, 2048);
  // 3) routing
  zero_f_kernel<<<1, 256, 0, stream>>>(ws + OFF_F);
  route_kernel<<<512, 256, 0, stream>>>(ws + OFF_ALLH, fqkQ, fqkK, fvw, gQ, gK,
                                        gV, fqk_emb, fv_emb, Wr_qk, Wr_v,
                                        ws + OFF_AQ, ws + OFF_AK, ws + OFF_AV,
                                        ws + OFF_F);
  // 4) restore GEMMs: Q/K/V = A(4096x512) * R(512x1024)
  gemm_f32_wmma<false><<<dim3(8, 32), 256, 0, stream>>>(
      ws + OFF_AQ, rneur, ws + OFF_Q, 4096, 1024, 512, 512, 1024, 1024);
  gemm_f32_wmma<false><<<dim3(8, 32), 256, 0, stream>>>(
      ws + OFF_AK, rneur, ws + OFF_K, 4096, 1024, 512, 512, 1024, 1024);
  gemm_f32_wmma<false><<<dim3(8, 32), 256, 0, stream>>>(
      ws + OFF_AV, rneur + (size_t)512 * 1024, ws + OFF_V, 4096, 1024, 512,
      512, 1024, 1024);
  // 5) q-norm scale
  qscale_kernel<<<512, 256, 0, stream>>>(ws + OFF_Q, ws + OFF_SCALE);
  // 6) causal flash attention (applies scale and 1/l)
  attn_kernel<<<dim3(32, 32), 256, 0, stream>>>(ws + OFF_Q, ws + OFF_K,
                                                ws + OFF_V, ws + OFF_SCALE,
                                                ws + OFF_ATTN);
  // 7) output projection: out = attn_out(4096x1024) * W_O^T
  gemm_f32_wmma<true><<<dim3(8, 32), 256, 0, stream>>>(
      ws + OFF_ATTN, W_O, out, 4096, 1024, 1024, 1024, 1024, 1024);
  // 8) aux scalar (last element of d_out)
  aux_kernel<<<1, 32, 0, stream>>>(ws + OFF_F, out + (out_size - 1));
}